// RecurrentOscillatorRNN_88158498717770
// MI455X (gfx1250) — compile-verified
//
#include <hip/hip_runtime.h>
#include <math.h>

// CDNA5 / gfx1250. Leaky RNN: latency-bound sequential scan; fp32 WMMA
// (throughput irrelevant at 134 MFLOP/step), W_hh pinned in LDS, h ping-pong
// in L2, per-batch-group global spin barrier.

typedef float v2f __attribute__((ext_vector_type(2)));
typedef float v8f __attribute__((ext_vector_type(8)));

#define B_    64
#define S_    4096
#define H_    1024
#define NB    16              // batch rows per group
#define NJ    16              // H columns per workgroup
#define NBG   (B_ / NB)       // 4 batch groups
#define NWGH  (H_ / NJ)       // 64 workgroups per batch group
#define HP    1028            // padded LDS row stride (floats); 1028 % 64 == 4
#define ALPHA 0.1f

// LDS partition sizes (floats)
#define SM_WS   (NJ * HP)               // W_hh slice   [16][1028]
#define SM_HP   (NB * HP)               // h tile       [16][1028]
#define SM_RED  (7 * 256)               // K-split reduction
#define SM_YRED 256
#define SM_TOTF (SM_WS + SM_HP + SM_RED + SM_YRED + NB + NJ + NJ)

__global__ void rnn_init_kernel(unsigned* cnt) {
  if (threadIdx.x < NBG) cnt[threadIdx.x] = 0u;
}

__global__ __launch_bounds__(256) void rnn_scan_kernel(
    const float* __restrict__ x,   const float* __restrict__ Whh,
    const float* __restrict__ Wxh, const float* __restrict__ bh,
    const float* __restrict__ Why, const float* __restrict__ by,
    float* __restrict__ out, float* __restrict__ hbuf,
    unsigned* __restrict__ cnt)
{
  extern __shared__ float smem[];
  float* Ws   = smem;                 // [NJ][HP]  W_hh rows j0..j0+15
  float* hp   = Ws   + SM_WS;         // [NB][HP]  current h rows (full H)
  float* red  = hp   + SM_HP;         // [7][256]  partial accumulators
  float* yred = red  + SM_RED;        // [256]
  float* xs   = yred + SM_YRED;       // [NB]
  float* wxs  = xs   + NB;            // [NJ]
  float* bhs  = wxs  + NJ;            // [NJ]

  const int tid  = threadIdx.x;
  const int lane = tid & 31;
  const int wave = tid >> 5;                 // 0..7 -> K-split slice
  const int bg   = blockIdx.x >> 6;          // 0..3
  const int hg   = blockIdx.x & 63;          // 0..63
  const int j0   = hg * NJ;
  const int b0   = bg * NB;

  // One-time: pin this WG's W_hh slice (and W_xh/b_h slice) in LDS.
  for (int idx = tid; idx < NJ * H_; idx += 256) {
    int j = idx >> 10, k = idx & (H_ - 1);
    Ws[j * HP + k] = Whh[(size_t)(j0 + j) * H_ + k];
  }
  if (tid < NJ) { wxs[tid] = Wxh[j0 + tid]; bhs[tid] = bh[j0 + tid]; }

  const int m    = lane & 15;                // fragment row/col within tile
  const int koff = (lane >> 4) << 1;         // 0 or 2 (K sub-offset per half-wave)
  const int k0   = wave * (H_ / 8);          // 128-wide K slice per wave
  unsigned* mycnt = cnt + bg;

  for (int t = 0; t < S_; ++t) {
    // ---- stage h(state after step t-1) into LDS; zeros at t==0 ----
    const float* hr = hbuf + (size_t)(t & 1) * (B_ * H_);
    for (int q = tid; q < NB * (H_ / 4); q += 256) {
      int b = q >> 8, k4 = q & 255;
      float4 v;
      if (t == 0) v = make_float4(0.f, 0.f, 0.f, 0.f);
      else        v = ((const float4*)(hr + (size_t)(b0 + b) * H_))[k4];
      *(float4*)(hp + b * HP + k4 * 4) = v;
    }
    if (tid < NB) xs[tid] = x[(size_t)(b0 + tid) * S_ + t];
    __syncthreads();

    // ---- y[t-1] from the freshly staged full h rows (one WG per bg) ----
    if (hg == 0 && t > 0) {
      int b = tid >> 4, seg = tid & 15;
      float s = 0.f;
      const float* hrow = hp + b * HP + seg * 64;
      #pragma unroll 8
      for (int k = 0; k < 64; ++k) s += hrow[k] * Why[seg * 64 + k];
      yred[tid] = s;
      __syncthreads();
      if (tid < NB) {
        float acc = by[0];
        for (int p = 0; p < 16; ++p) acc += yred[tid * 16 + p];
        out[(size_t)(b0 + tid) * S_ + (t - 1)] = acc;
      }
      __syncthreads();
    }

    // ---- [16b x 16j] tile: fp32 WMMA, K=1024 split 8 ways over waves ----
    v8f c = {0.f, 0.f, 0.f, 0.f, 0.f, 0.f, 0.f, 0.f};
    const float* ha = hp + m * HP + k0 + koff;   // A: h[b=m][k..k+1]
    const float* wb = Ws + m * HP + k0 + koff;   // B: Whh[j=m][k..k+1]
    #pragma unroll 8
    for (int kk = 0; kk < H_ / 8; kk += 4) {
      v2f a = *(const v2f*)(ha + kk);
      v2f b = *(const v2f*)(wb + kk);
      c = __builtin_amdgcn_wmma_f32_16x16x4_f32(false, a, false, b,
                                                (short)0, c, false, false);
    }

    // ---- reduce the 8 K-partials through LDS ----
    if (wave > 0) {
      #pragma unroll
      for (int r = 0; r < 8; ++r) red[(wave - 1) * 256 + r * 32 + lane] = c[r];
    }
    __syncthreads();
    if (wave == 0) {
      #pragma unroll
      for (int p = 0; p < 7; ++p)
        #pragma unroll
        for (int r = 0; r < 8; ++r) c[r] += red[p * 256 + r * 32 + lane];
      // ---- epilogue: tanh + leaky update; write next-state slice ----
      float* hw = hbuf + (size_t)((t + 1) & 1) * (B_ * H_);
      #pragma unroll
      for (int r = 0; r < 8; ++r) {
        int b = (lane < 16) ? r : (r + 8);     // C/D layout: M=r (+8 hi half)
        float pre = c[r] + xs[b] * wxs[m] + bhs[m];
        float hn  = tanhf(pre);
        float ho  = (1.0f - ALPHA) * hp[b * HP + j0 + m] + ALPHA * hn;
        hw[(size_t)(b0 + b) * H_ + j0 + m] = ho;
        if (t == S_ - 1)
          out[(size_t)B_ * S_ + (size_t)(b0 + b) * H_ + j0 + m] = ho;
      }
    }

    // ---- batch-group barrier: 64 WGs, monotonic arrival counter ----
    __threadfence();
    __syncthreads();
    if (tid == 0) {
      __hip_atomic_fetch_add(mycnt, 1u, __ATOMIC_RELEASE,
                             __HIP_MEMORY_SCOPE_AGENT);
      unsigned target = 64u * (unsigned)(t + 1);
      while (__hip_atomic_load(mycnt, __ATOMIC_ACQUIRE,
                               __HIP_MEMORY_SCOPE_AGENT) < target)
        __builtin_amdgcn_s_sleep(2);
    }
    __syncthreads();
    __threadfence();
  }

  // ---- final y[S-1] from state after the last step (buffer S_&1 == 0) ----
  if (hg == 0) {
    const float* hr = hbuf + (size_t)(S_ & 1) * (B_ * H_);
    for (int q = tid; q < NB * (H_ / 4); q += 256) {
      int b = q >> 8, k4 = q & 255;
      *(float4*)(hp + b * HP + k4 * 4) =
          ((const float4*)(hr + (size_t)(b0 + b) * H_))[k4];
    }
    __syncthreads();
    int b = tid >> 4, seg = tid & 15;
    float s = 0.f;
    const float* hrow = hp + b * HP + seg * 64;
    #pragma unroll 8
    for (int k = 0; k < 64; ++k) s += hrow[k] * Why[seg * 64 + k];
    yred[tid] = s;
    __syncthreads();
    if (tid < NB) {
      float acc = by[0];
      for (int p = 0; p < 16; ++p) acc += yred[tid * 16 + p];
      out[(size_t)(b0 + tid) * S_ + (S_ - 1)] = acc;
    }
  }
}

extern "C" void kernel_launch(void* const* d_in, const int* in_sizes, int n_in,
                              void* d_out, int out_size, void* d_ws, size_t ws_size,
                              hipStream_t stream) {
  (void)in_sizes; (void)n_in; (void)out_size; (void)ws_size;
  const float* x   = (const float*)d_in[0];  // [64,4096,1]
  const float* Whh = (const float*)d_in[1];  // [1024,1024]
  const float* Wxh = (const float*)d_in[2];  // [1024,1]
  const float* bh  = (const float*)d_in[3];  // [1024]
  const float* Why = (const float*)d_in[4];  // [1,1024]
  const float* by  = (const float*)d_in[5];  // [1]
  float* out = (float*)d_out;                // [64*4096] outputs ++ [64*1024] h_final

  // Workspace: two h ping-pong buffers (2*64*1024 f32) + 4 arrival counters.
  float*    hbuf = (float*)d_ws;
  unsigned* cnt  = (unsigned*)((char*)d_ws + (size_t)2 * B_ * H_ * sizeof(float));

  const size_t smem_bytes = (size_t)SM_TOTF * sizeof(float);  // ~137 KB
  (void)hipFuncSetAttribute((const void*)rnn_scan_kernel,
                            hipFuncAttributeMaxDynamicSharedMemorySize,
                            (int)smem_bytes);

  rnn_init_kernel<<<1, 64, 0, stream>>>(cnt);
  rnn_scan_kernel<<<NBG * NWGH, 256, smem_bytes, stream>>>(
      x, Whh, Wxh, bh, Why, by, out, hbuf, cnt);
}